// DiffusionCriterion_27092653703623
// MI455X (gfx1250) — compile-verified
//
#include <hip/hip_runtime.h>
#include <math.h>

// Problem constants (from reference setup_inputs)
#define B_  128
#define Q_  500
#define C_  256
#define T_  100
#define QT_ (Q_ * T_)

typedef float v2f __attribute__((ext_vector_type(2)));
typedef float v8f __attribute__((ext_vector_type(8)));

// ---------------------------------------------------------------------------
// Kernel 1: per-row softmax stats (rowmax, sum of exp). One wave32 per row.
// ---------------------------------------------------------------------------
__global__ void rowstats_kernel(const float* __restrict__ logits,
                                float* __restrict__ rowmax,
                                float* __restrict__ rowsum) {
    const int wid  = threadIdx.x >> 5;       // 8 waves / block
    const int lane = threadIdx.x & 31;
    const int row  = blockIdx.x * 8 + wid;   // row in [0, B*Q)
    if (row >= B_ * Q_) return;
    const float* p = logits + (size_t)row * C_;

    float v[8];
    float m = -INFINITY;
#pragma unroll
    for (int j = 0; j < 8; ++j) {            // 256 = 32 lanes * 8
        v[j] = p[lane + 32 * j];
        m = fmaxf(m, v[j]);
    }
#pragma unroll
    for (int off = 16; off; off >>= 1) m = fmaxf(m, __shfl_xor(m, off, 32));

    float s = 0.0f;
#pragma unroll
    for (int j = 0; j < 8; ++j) s += __expf(v[j] - m);
#pragma unroll
    for (int off = 16; off; off >>= 1) s += __shfl_xor(s, off, 32);

    if (lane == 0) { rowmax[row] = m; rowsum[row] = s; }
}

// ---------------------------------------------------------------------------
// Kernel 2: cost matrix. One wave per 16x16 (q,t) tile.
// cost_class via f32 WMMA against a register one-hot (exact gather).
// ---------------------------------------------------------------------------
__global__ void cost_kernel(const float* __restrict__ logits,
                            const float* __restrict__ boxes,
                            const int*   __restrict__ labels,
                            const float* __restrict__ tboxes,
                            const float* __restrict__ rowmax,
                            const float* __restrict__ rowsum,
                            float* __restrict__ cost) {
    const int lane = threadIdx.x;            // blockDim.x == 32
    const int t0   = blockIdx.x * 16;        // 7 tiles (T padded to 112)
    const int q0   = blockIdx.y * 16;        // 32 tiles (Q padded to 512)
    const int b    = blockIdx.z;
    const int half = lane >> 4;              // 0 | 1
    const int lidx = lane & 15;

    // A-matrix lane row (clamped; out-of-range rows are discarded at store)
    const int qa = min(q0 + lidx, Q_ - 1);
    const float* lrow = logits + ((size_t)b * Q_ + qa) * C_;
    const float  am   = rowmax[b * Q_ + qa];
    const float  ainv = 1.0f / rowsum[b * Q_ + qa];

    // B-matrix lane column: one-hot of labels[t0+lidx]
    const int tcol = t0 + lidx;
    const int lbl  = (tcol < T_) ? labels[b * T_ + tcol] : -1;

    v8f acc = {};
    for (int kb = 0; kb < C_; kb += 4) {
        const int k0 = kb + 2 * half;        // ISA layout: K = 2*half + vgpr
        v2f a, bm;
        a.x  = __expf(lrow[k0]     - am) * ainv;
        a.y  = __expf(lrow[k0 + 1] - am) * ainv;
        bm.x = (lbl == k0)     ? 1.0f : 0.0f;
        bm.y = (lbl == k0 + 1) ? 1.0f : 0.0f;
        // D = A(16x4) * B(4x16) + C ; exact: one-hot -> exact gather of prob
        acc = __builtin_amdgcn_wmma_f32_16x16x4_f32(
            false, a, false, bm, (short)0, acc, false, false);
    }
    // acc[j] = prob[q0 + j + 8*half][t0 + lidx]

    // target box (clamped load; guarded store)
    const int tcl = min(tcol, T_ - 1);
    const float4 tb4 = *(const float4*)(tboxes + ((size_t)b * T_ + tcl) * 4);
    const float tx1 = tb4.x - 0.5f * tb4.z, ty1 = tb4.y - 0.5f * tb4.w;
    const float tx2 = tb4.x + 0.5f * tb4.z, ty2 = tb4.y + 0.5f * tb4.w;
    const float tarea = (tx2 - tx1) * (ty2 - ty1);

    float* cbase = cost + (size_t)b * QT_;
#pragma unroll
    for (int j = 0; j < 8; ++j) {
        const int q  = q0 + j + 8 * half;
        const int qc = min(q, Q_ - 1);
        const float4 pb4 = *(const float4*)(boxes + ((size_t)b * Q_ + qc) * 4);

        const float l1 = fabsf(pb4.x - tb4.x) + fabsf(pb4.y - tb4.y) +
                         fabsf(pb4.z - tb4.z) + fabsf(pb4.w - tb4.w);

        const float px1 = pb4.x - 0.5f * pb4.z, py1 = pb4.y - 0.5f * pb4.w;
        const float px2 = pb4.x + 0.5f * pb4.z, py2 = pb4.y + 0.5f * pb4.w;
        const float parea = (px2 - px1) * (py2 - py1);

        const float ix1 = fmaxf(px1, tx1), iy1 = fmaxf(py1, ty1);
        const float ix2 = fminf(px2, tx2), iy2 = fminf(py2, ty2);
        const float inter = fmaxf(ix2 - ix1, 0.0f) * fmaxf(iy2 - iy1, 0.0f);
        const float uni   = parea + tarea - inter;
        const float iou   = inter / uni;

        const float ex1 = fminf(px1, tx1), ey1 = fminf(py1, ty1);
        const float ex2 = fmaxf(px2, tx2), ey2 = fmaxf(py2, ty2);
        const float earea = fmaxf(ex2 - ex1, 0.0f) * fmaxf(ey2 - ey1, 0.0f);
        const float giou  = iou - (earea - uni) / earea;

        const float cval = -acc[j] + 5.0f * l1 - 2.0f * giou;
        if (q < Q_ && tcol < T_) cbase[q * T_ + tcol] = cval;  // flat == ref
    }
}

// ---------------------------------------------------------------------------
// Kernel 3: greedy matching (100 sequential block-wide argmins) + losses.
// One 1024-thread block per image; cost matrix is L2-resident.
// ---------------------------------------------------------------------------
__device__ __forceinline__ void amin2(float& v, int& i, float ov, int oi) {
    if (ov < v || (ov == v && oi < i)) { v = ov; i = oi; }
}

__global__ __launch_bounds__(1024)
void match_loss_kernel(const float* __restrict__ logits,
                       const float* __restrict__ boxes,
                       const int*   __restrict__ labels,
                       const float* __restrict__ tboxes,
                       const int*   __restrict__ ttime,
                       const float* __restrict__ noise,
                       const float* __restrict__ rowmax,
                       const float* __restrict__ rowsum,
                       const float* __restrict__ cost,
                       float* __restrict__ per_img) {
    const int b    = blockIdx.x;
    const int tid  = threadIdx.x;
    const int lane = tid & 31;
    const int wid  = tid >> 5;
    const int nw   = blockDim.x >> 5;

    __shared__ int   row_used[Q_];
    __shared__ int   col_used[T_];
    __shared__ int   s_src[T_], s_tgt[T_];
    __shared__ float s_v[32];
    __shared__ int   s_i[32];
    __shared__ float s_scale;
    __shared__ float s_acc[4];

    for (int i = tid; i < Q_; i += blockDim.x) row_used[i] = 0;
    for (int i = tid; i < T_; i += blockDim.x) col_used[i] = 0;
    if (tid < 4) s_acc[tid] = 0.0f;
    __syncthreads();

    const float* Cm = cost + (size_t)b * QT_;

    for (int step = 0; step < T_; ++step) {
        float bv = INFINITY;
        int   bi = 0x7FFFFFFF;
        for (int idx = tid; idx < QT_; idx += blockDim.x) {
            const int q = idx / T_;
            const int t = idx - q * T_;
            if (row_used[q] | col_used[t]) continue;
            const float v = Cm[idx];
            amin2(bv, bi, v, idx);
        }
        for (int off = 16; off; off >>= 1)
            amin2(bv, bi, __shfl_xor(bv, off, 32), __shfl_xor(bi, off, 32));
        if (lane == 0) { s_v[wid] = bv; s_i[wid] = bi; }
        __syncthreads();
        if (wid == 0) {
            bv = (lane < nw) ? s_v[lane] : INFINITY;
            bi = (lane < nw) ? s_i[lane] : 0x7FFFFFFF;
            for (int off = 16; off; off >>= 1)
                amin2(bv, bi, __shfl_xor(bv, off, 32), __shfl_xor(bi, off, 32));
            if (lane == 0) {
                const int q = bi / T_;
                const int t = bi - q * T_;
                row_used[q] = 1;
                col_used[t] = 1;
                s_src[step] = q;
                s_tgt[step] = t;
            }
        }
        __syncthreads();
    }

    // diffusion noise scale: acp = cumprod(1 - linspace(1e-4, 0.02, 100))[t]
    if (tid == 0) {
        const int tt = ttime[b];
        float acp = 1.0f;
        for (int i = 0; i <= tt; ++i) {
            const float beta = 1e-4f + (0.02f - 1e-4f) * ((float)i / 99.0f);
            acp *= (1.0f - beta);
        }
        s_scale = sqrtf(1.0f - acp);
    }
    __syncthreads();

    if (tid < T_) {
        const int sq = s_src[tid], st = s_tgt[tid];
        const int cls = labels[b * T_ + st];

        const float lg = logits[((size_t)b * Q_ + sq) * C_ + cls];
        const float ce = -(lg - rowmax[b * Q_ + sq] - logf(rowsum[b * Q_ + sq]));

        const float4 sb = *(const float4*)(boxes  + ((size_t)b * Q_ + sq) * 4);
        const float4 gb = *(const float4*)(tboxes + ((size_t)b * T_ + st) * 4);
        const float4 nz = *(const float4*)(noise  + ((size_t)b * T_ + st) * 4);

        const float l1 = fabsf(sb.x - gb.x) + fabsf(sb.y - gb.y) +
                         fabsf(sb.z - gb.z) + fabsf(sb.w - gb.w);

        // elementwise GIoU on xyxy
        const float px1 = sb.x - 0.5f * sb.z, py1 = sb.y - 0.5f * sb.w;
        const float px2 = sb.x + 0.5f * sb.z, py2 = sb.y + 0.5f * sb.w;
        const float gx1 = gb.x - 0.5f * gb.z, gy1 = gb.y - 0.5f * gb.w;
        const float gx2 = gb.x + 0.5f * gb.z, gy2 = gb.y + 0.5f * gb.w;
        const float a1 = (px2 - px1) * (py2 - py1);
        const float a2 = (gx2 - gx1) * (gy2 - gy1);
        const float inter = fmaxf(fminf(px2, gx2) - fmaxf(px1, gx1), 0.0f) *
                            fmaxf(fminf(py2, gy2) - fmaxf(py1, gy1), 0.0f);
        const float uni = a1 + a2 - inter;
        const float ea  = fmaxf(fmaxf(px2, gx2) - fminf(px1, gx1), 0.0f) *
                          fmaxf(fmaxf(py2, gy2) - fminf(py1, gy1), 0.0f);
        const float giou = inter / uni - (ea - uni) / ea;

        const float ns = s_scale;
        const float dx = sb.x - (gb.x + nz.x * ns);
        const float dy = sb.y - (gb.y + nz.y * ns);
        const float dz = sb.z - (gb.z + nz.z * ns);
        const float dw = sb.w - (gb.w + nz.w * ns);
        const float sq2 = dx * dx + dy * dy + dz * dz + dw * dw;

        atomicAdd(&s_acc[0], ce);
        atomicAdd(&s_acc[1], l1);
        atomicAdd(&s_acc[2], giou);
        atomicAdd(&s_acc[3], sq2);
    }
    __syncthreads();

    if (tid == 0) {
        const float per = 1.0f * (s_acc[0] / (float)T_)
                        + 5.0f * (s_acc[1] / (float)T_)
                        + 2.0f * (1.0f - s_acc[2] / (float)T_)
                        + 1.0f * (s_acc[3] / (float)(T_ * 4));
        per_img[b] = per;
    }
}

// ---------------------------------------------------------------------------
// Kernel 4: deterministic batch mean.
// ---------------------------------------------------------------------------
__global__ void finalize_kernel(const float* __restrict__ per_img,
                                float* __restrict__ out) {
    if (threadIdx.x == 0) {
        float s = 0.0f;
        for (int i = 0; i < B_; ++i) s += per_img[i];
        out[0] = s / (float)B_;
    }
}

// ---------------------------------------------------------------------------
extern "C" void kernel_launch(void* const* d_in, const int* in_sizes, int n_in,
                              void* d_out, int out_size, void* d_ws, size_t ws_size,
                              hipStream_t stream) {
    const float* logits = (const float*)d_in[0];  // [B,Q,C]
    const float* boxes  = (const float*)d_in[1];  // [B,Q,4]
    const int*   labels = (const int*)  d_in[2];  // [B,T]
    const float* tboxes = (const float*)d_in[3];  // [B,T,4]
    const int*   ttime  = (const int*)  d_in[4];  // [B]
    const float* noise  = (const float*)d_in[5];  // [B,T,4]
    float* out = (float*)d_out;

    float* ws      = (float*)d_ws;
    float* rowmax  = ws;                                   // B*Q
    float* rowsum  = ws + (size_t)B_ * Q_;                 // B*Q
    float* cost    = ws + 2 * (size_t)B_ * Q_;             // B*Q*T
    float* per_img = cost + (size_t)B_ * QT_;              // B

    rowstats_kernel<<<(B_ * Q_) / 8, 256, 0, stream>>>(logits, rowmax, rowsum);

    dim3 gc(7, 32, B_);  // t-tiles (112/16), q-tiles (512/16), batch
    cost_kernel<<<gc, 32, 0, stream>>>(logits, boxes, labels, tboxes,
                                       rowmax, rowsum, cost);

    match_loss_kernel<<<B_, 1024, 0, stream>>>(logits, boxes, labels, tboxes,
                                               ttime, noise, rowmax, rowsum,
                                               cost, per_img);

    finalize_kernel<<<1, 32, 0, stream>>>(per_img, out);
}